// DecayedVoteAssociativeLM_34368328303151
// MI455X (gfx1250) — compile-verified
//
#include <hip/hip_runtime.h>
#include <math.h>

// Problem constants (from reference): V=50257, E=256, H=512, B=4, S=512
#define VV 50257
#define EE 256
#define HH 512
#define BB 4
#define SS 512
#define GH (3*HH)          // 1536
#define MROWS (BB*SS)      // 2048

typedef float v2f __attribute__((ext_vector_type(2)));
typedef float v8f __attribute__((ext_vector_type(8)));
typedef unsigned tdm_g0_t __attribute__((ext_vector_type(4)));
typedef unsigned tdm_g1_t __attribute__((ext_vector_type(8)));

// Flip to 0 to fall back to the round-1 synchronous LDS fill if the TDM asm
// form needs adjustment.
#define K4_USE_TDM 1

__device__ __forceinline__ float sigmf(float x) { return 1.0f / (1.0f + expf(-x)); }

// ---------------------------------------------------------------------------
// K0: zero h ping-pong buffers and grid-sync counters (ws is poisoned 0xAA)
// ---------------------------------------------------------------------------
__global__ void k0_init(float* __restrict__ hbuf, int* __restrict__ cnt) {
    int t = blockIdx.x * blockDim.x + threadIdx.x;
    if (t < 2 * BB * HH) hbuf[t] = 0.0f;    // hbuf[2][B][H]
    if (t < BB) cnt[t] = 0;
}

// ---------------------------------------------------------------------------
// K1: xg[row, g] = dot(embedding[ids[row]], W_ih[g,:]) + b_ih[g]
// ---------------------------------------------------------------------------
__global__ __launch_bounds__(256) void k1_xg(const int* __restrict__ ids,
                                             const float* __restrict__ emb,
                                             const float* __restrict__ W_ih,
                                             const float* __restrict__ b_ih,
                                             float* __restrict__ xg) {
    __shared__ float e[EE];
    const int row = blockIdx.x;
    const int t = threadIdx.x;
    const int id = ids[row];
    e[t] = emb[(size_t)id * EE + t];
    __syncthreads();
    for (int u = 0; u < GH / 256; ++u) {
        const int g = t + 256 * u;
        const float* wr = W_ih + (size_t)g * EE;
        float acc = b_ih[g];
#pragma unroll 8
        for (int k = 0; k < EE; ++k) acc += e[k] * wr[k];
        xg[(size_t)row * GH + g] = acc;
    }
}

// ---------------------------------------------------------------------------
// K2: GRU scan. 64 WGs = 4 batches x 16 H-chunks (32 channels each).
// W_hh slab (192KB) cached in LDS, 516-float pitch (bank-conflict-free
// strided-K, 4 threads/row + __shfl_xor quad reduce). Per-step grid sync
// among the 16 WGs of a batch via agent-scope atomics; h double-buffered.
// ---------------------------------------------------------------------------
#define K2_WPITCH 516
#define K2_LDS_FLOATS (96 * K2_WPITCH + HH + 96)
#define K2_LDS_BYTES (K2_LDS_FLOATS * 4)

__global__ __launch_bounds__(384) void k2_gru(const float* __restrict__ xg,
                                              const float* __restrict__ W_hh,
                                              const float* __restrict__ b_hh,
                                              float* __restrict__ states,
                                              float* __restrict__ hbuf,
                                              int* __restrict__ cnt) {
    extern __shared__ float lds[];
    float* Wl  = lds;                      // [96][516]
    float* hld = lds + 96 * K2_WPITCH;     // [512]
    float* hg  = hld + HH;                 // [96]

    const int b  = blockIdx.x >> 4;
    const int c0 = (blockIdx.x & 15) * 32;
    const int t  = threadIdx.x;

    for (int i = 0; i < (96 * HH) / 384; ++i) {
        const int li = t + 384 * i;
        const int g = li >> 9, k = li & 511;
        const int gate = g >> 5, ch = g & 31;
        Wl[g * K2_WPITCH + k] = W_hh[((size_t)(gate * HH + c0 + ch)) * HH + k];
    }
    __syncthreads();

    const int g = t >> 2, p = t & 3;
    for (int s = 0; s < SS; ++s) {
        if (s > 0) {
            if (t == 0) {
                const int target = 16 * s;
                while (__hip_atomic_load(&cnt[b], __ATOMIC_ACQUIRE,
                                         __HIP_MEMORY_SCOPE_AGENT) < target)
                    __builtin_amdgcn_s_sleep(1);
            }
            __syncthreads();
        }
        const float* hsrc = hbuf + (size_t)(s & 1) * (BB * HH) + b * HH;
        for (int idx = t; idx < HH; idx += 384) hld[idx] = hsrc[idx];
        __syncthreads();

        float acc = 0.0f;
        const float* wrow = Wl + g * K2_WPITCH + p;
#pragma unroll 8
        for (int j = 0; j < HH / 4; ++j) acc += wrow[4 * j] * hld[4 * j + p];
        acc += __shfl_xor(acc, 1);
        acc += __shfl_xor(acc, 2);
        if (p == 0) hg[g] = acc;
        __syncthreads();

        if (t < 32) {
            const int c = c0 + t;
            const size_t xrow = (size_t)(b * SS + s) * GH;
            const float hr = hg[t]      + b_hh[c];
            const float hz = hg[32 + t] + b_hh[HH + c];
            const float hn = hg[64 + t] + b_hh[2 * HH + c];
            const float r = sigmf(xg[xrow + c] + hr);
            const float z = sigmf(xg[xrow + HH + c] + hz);
            const float n = tanhf(xg[xrow + 2 * HH + c] + r * hn);
            const float hnew = (1.0f - z) * n + z * hld[c];
            states[(size_t)(b * SS + s) * HH + c] = hnew;
            hbuf[(size_t)((s + 1) & 1) * (BB * HH) + b * HH + c] = hnew;
        }
        __syncthreads();
        if (t == 0) {
            __threadfence();
            __hip_atomic_fetch_add(&cnt[b], 1, __ATOMIC_RELEASE,
                                   __HIP_MEMORY_SCOPE_AGENT);
        }
    }
}

// ---------------------------------------------------------------------------
// K3: proj[row, e] = dot(states[row,:], W_he[e,:]) + b_he[e]
// ---------------------------------------------------------------------------
__global__ __launch_bounds__(256) void k3_proj(const float* __restrict__ states,
                                               const float* __restrict__ W_he,
                                               const float* __restrict__ b_he,
                                               float* __restrict__ proj) {
    __shared__ float hrow[HH];
    const int row = blockIdx.x;
    const int t = threadIdx.x;
    const float* src = states + (size_t)row * HH;
    hrow[t] = src[t];
    hrow[t + 256] = src[t + 256];
    __syncthreads();
    const float* wr = W_he + (size_t)t * HH;
    float acc = b_he[t];
#pragma unroll 8
    for (int k = 0; k < HH; ++k) acc += hrow[k] * wr[k];
    proj[(size_t)row * EE + t] = acc;
}

// ---------------------------------------------------------------------------
// K4: logits GEMM with V_WMMA_F32_16X16X4_F32 + TDM double-buffered B tiles.
// out[row, v] = dot(proj[row,:], embedding[v,:]) + output_bias[v]
// WG(256 thr = 8 waves): 128(M) x 128(N); wave w owns rows [16w,16w+16) with
// 8 f32 16x16 accumulators. B tile = embedding[n0..n0+127][kc..kc+31] staged
// by TENSOR_LOAD_TO_LDS (wave 0 issues; TENSORcnt tracks; D# pad fields
// produce the 34-float LDS pitch; OOB rows beyond V return zeros).
// ---------------------------------------------------------------------------
#define T_N 128
#define T_M 128
#define KC 32
#define BPAD 34

#if K4_USE_TDM
__device__ __forceinline__ void tdm_issue_tile(const float* gbase,
                                               unsigned lds_byte_addr,
                                               unsigned rows_remaining) {
    const unsigned long long ga = (unsigned long long)(uintptr_t)gbase;
    tdm_g0_t g0;
    g0[0] = 1u;                                   // count=1, user-mode load
    g0[1] = lds_byte_addr;                        // lds_addr [63:32]
    g0[2] = (unsigned)(ga & 0xffffffffu);         // global_addr lo
    g0[3] = (unsigned)((ga >> 32) & 0x01ffffffu)  // global_addr [56:32]
          | (2u << 30);                           // type = 2 (image/tensor)
    tdm_g1_t g1;
    g1[0] = (2u << 16)                            // data_size = 4 bytes
          | (1u << 20)                            // pad_enable
          | (4u << 22)                            // pad_interval: 32 DWORDs
          | (1u << 25);                           // pad_amount:   2 DWORDs
    g1[1] = (256u << 16);                         // tensor_dim0 = 256 (lo16)
    g1[2] = (rows_remaining & 0xffffu) << 16;     // tensor_dim1 lo16
    g1[3] = ((rows_remaining >> 16) & 0xffffu)    // tensor_dim1 hi16
          | ((unsigned)KC << 16);                 // tile_dim0 = 32
    g1[4] = (unsigned)T_N;                        // tile_dim1 = 128, tile_dim2=0
    g1[5] = 256u;                                 // tensor_dim0_stride lo32
    g1[6] = 0u;                                   // stride hi | dim1_stride lo
    g1[7] = 0u;
    asm volatile("tensor_load_to_lds %0, %1" :: "s"(g0), "s"(g1) : "memory");
}
#endif

__global__ __launch_bounds__(256) void k4_logits(const float* __restrict__ proj,
                                                 const float* __restrict__ emb,
                                                 const float* __restrict__ obias,
                                                 float* __restrict__ out) {
    __shared__ float Bt[2][T_N * BPAD];
    const int n0 = blockIdx.x * T_N;
    const int m0 = blockIdx.y * T_M;
    const int t = threadIdx.x;
    const int wave = t >> 5;
    const int lane = t & 31;
    const int lo = lane & 15;
    const int hi = lane >> 4;

    v8f acc[8] = {};
    const float* Arow = proj + (size_t)(m0 + wave * 16 + lo) * EE;

#if K4_USE_TDM
    const unsigned rows_rem = (unsigned)(VV - n0);
    if (wave == 0) {
        tdm_issue_tile(emb + (size_t)n0 * EE,
                       (unsigned)(uintptr_t)(void*)&Bt[0][0], rows_rem);
    }
#endif

    for (int kc = 0; kc < EE; kc += KC) {
        const int cur = (kc >> 5) & 1;
        const bool has_next = (kc + KC) < EE;

#if K4_USE_TDM
        if (wave == 0) {
            if (has_next) {
                // Bt[cur^1] was last read in the previous iteration; safe to refill.
                tdm_issue_tile(emb + (size_t)n0 * EE + (kc + KC),
                               (unsigned)(uintptr_t)(void*)&Bt[cur ^ 1][0],
                               rows_rem);
                __builtin_amdgcn_s_wait_tensorcnt(1);  // fill(cur) complete
            } else {
                __builtin_amdgcn_s_wait_tensorcnt(0);
            }
        }
        __syncthreads();
#else
        __syncthreads();
#pragma unroll
        for (int i = 0; i < 8; ++i) {
            const int li = t + 256 * i;
            const int n = li >> 4;
            const int kp = li & 15;
            int col = n0 + n;
            if (col >= VV) col = VV - 1;
            const float* src = emb + (size_t)col * EE + kc + kp * 2;
            *(v2f*)&Bt[cur][n * BPAD + kp * 2] = *(const v2f*)src;
        }
        __syncthreads();
#endif

        if (has_next) __builtin_prefetch(Arow + kc + KC, 0, 1);

#pragma unroll
        for (int kk = 0; kk < KC; kk += 4) {
            const v2f a = *(const v2f*)(Arow + kc + kk + 2 * hi);
#pragma unroll
            for (int nt = 0; nt < 8; ++nt) {
                const v2f bf =
                    *(const v2f*)&Bt[cur][(nt * 16 + lo) * BPAD + kk + 2 * hi];
                acc[nt] = __builtin_amdgcn_wmma_f32_16x16x4_f32(
                    false, a, false, bf, (short)0, acc[nt], false, false);
            }
        }
        __syncthreads();   // all reads of Bt[cur] done before it is refilled
    }

#pragma unroll
    for (int nt = 0; nt < 8; ++nt) {
        const int col = n0 + nt * 16 + lo;
        if (col < VV) {
            const float bias = obias[col];
#pragma unroll
            for (int r = 0; r < 8; ++r) {
                const int row = m0 + wave * 16 + r + 8 * hi;
                out[(size_t)row * VV + col] = acc[nt][r] + bias;
            }
        }
    }
}

// ---------------------------------------------------------------------------
// K5: gates + sparse memory-scan correction.
// m_s has <= s nonzeros: coefficient of token t is write_t * prod decay.
// Thread t carries c_t; scatter-adds read_s * c_t at (s, ids[t]) via f32 atomics.
// ---------------------------------------------------------------------------
__global__ __launch_bounds__(512) void k5_corr(const float* __restrict__ states,
                                               const int* __restrict__ ids,
                                               const float* __restrict__ W_read,
                                               const float* __restrict__ b_read,
                                               const float* __restrict__ W_write,
                                               const float* __restrict__ b_write,
                                               const float* __restrict__ W_decay,
                                               const float* __restrict__ b_decay,
                                               const float* __restrict__ mscale,
                                               float* __restrict__ out) {
    __shared__ float rd[SS], dc[SS], wr[SS];
    __shared__ int il[SS];
    const int b = blockIdx.x;
    const int t = threadIdx.x;

    const float* h = states + (size_t)(b * SS + t) * HH;
    float ar = b_read[0], ad = b_decay[0], aw = b_write[0];
#pragma unroll 8
    for (int k = 0; k < HH; ++k) {
        const float hv = h[k];
        ar += hv * W_read[k];
        ad += hv * W_decay[k];
        aw += hv * W_write[k];
    }
    rd[t] = sigmf(ar) * mscale[0];
    dc[t] = sigmf(ad);
    wr[t] = sigmf(aw);
    il[t] = ids[b * SS + t];
    __syncthreads();

    float coeff = 0.0f;
    const int myid = il[t];
    const size_t obase = (size_t)b * SS * (size_t)VV;
    for (int s = 0; s < SS; ++s) {
        if (t < s) {
            atomicAdd(&out[obase + (size_t)s * VV + myid], rd[s] * coeff);
            coeff *= dc[s];
        } else if (t == s) {
            coeff = wr[s];
        }
    }
}

// ---------------------------------------------------------------------------
// launch
// ---------------------------------------------------------------------------
extern "C" void kernel_launch(void* const* d_in, const int* in_sizes, int n_in,
                              void* d_out, int out_size, void* d_ws, size_t ws_size,
                              hipStream_t stream) {
    (void)in_sizes; (void)n_in; (void)out_size; (void)ws_size;
    const int*   ids     = (const int*)d_in[0];
    const float* emb     = (const float*)d_in[1];
    const float* W_ih    = (const float*)d_in[2];
    const float* W_hh    = (const float*)d_in[3];
    const float* b_ih    = (const float*)d_in[4];
    const float* b_hh    = (const float*)d_in[5];
    const float* W_he    = (const float*)d_in[6];
    const float* b_he    = (const float*)d_in[7];
    const float* obias   = (const float*)d_in[8];
    const float* W_read  = (const float*)d_in[9];
    const float* b_read  = (const float*)d_in[10];
    const float* W_write = (const float*)d_in[11];
    const float* b_write = (const float*)d_in[12];
    const float* W_decay = (const float*)d_in[13];
    const float* b_decay = (const float*)d_in[14];
    const float* mscale  = (const float*)d_in[15];
    float* out = (float*)d_out;

    char* w = (char*)d_ws;
    float* xg     = (float*)(w);                 // B*S*3H f32 = 12,582,912 B
    float* states = (float*)(w + 12582912);      // B*S*H    =  4,194,304 B
    float* proj   = (float*)(w + 16777216);      // B*S*E    =  2,097,152 B
    float* hbuf   = (float*)(w + 18874368);      // 2*B*H    =     16,384 B
    int*   cnt    = (int*)  (w + 18890752);      // B counters

    (void)hipFuncSetAttribute((const void*)k2_gru,
                              hipFuncAttributeMaxDynamicSharedMemorySize,
                              K2_LDS_BYTES);

    k0_init<<<16, 256, 0, stream>>>(hbuf, cnt);
    k1_xg<<<MROWS, 256, 0, stream>>>(ids, emb, W_ih, b_ih, xg);
    k2_gru<<<BB * 16, 384, K2_LDS_BYTES, stream>>>(xg, W_hh, b_hh, states, hbuf, cnt);
    k3_proj<<<MROWS, 256, 0, stream>>>(states, W_he, b_he, proj);
    k4_logits<<<dim3((VV + T_N - 1) / T_N, MROWS / T_M), 256, 0, stream>>>(proj, emb, obias, out);
    k5_corr<<<BB, 512, 0, stream>>>(states, ids, W_read, b_read, W_write, b_write,
                                    W_decay, b_decay, mscale, out);
}